// CostVolume_26216480375278
// MI455X (gfx1250) — compile-verified
//
#include <hip/hip_runtime.h>

// CDNA5 (gfx1250) wave32 WMMA group-correlation cost volume.
// cost[b,g,h,w,d] = (1/8) * sum_c L[b,g*8+c,h,w] * R[b,g*8+c,h,w-d], 0 if w<d.
// Banded 320x320xK8 matmul per (b,h,g) slice via V_WMMA_F32_16X16X4_F32;
// diagonal band re-indexed through per-wave LDS staging so global stores are
// dense 512B-contiguous b128 NT stores (kernel is HBM-store-bound: ~524MB
// traffic -> ~22.5us floor at 23.3 TB/s).

typedef float v2f __attribute__((ext_vector_type(2)));
typedef float v4f __attribute__((ext_vector_type(4)));
typedef float v8f __attribute__((ext_vector_type(8)));

#define NB     4    // batch
#define NC     64   // channels
#define NG     8    // groups
#define CGRP   8    // channels per group
#define NH     160
#define NW     320
#define ND     64
#define LDSW   336  // input staging row stride (2*336 mod 64 == 32: conflict-free halves)
#define STW    68   // output staging row stride: 64 data + 4 dump/pad slots
#define NWAVES 8

__global__ __launch_bounds__(256) void cost_volume_wmma_kernel(
    const float* __restrict__ left, const float* __restrict__ right,
    float* __restrict__ out)
{
    __shared__ alignas(16) float Ls[CGRP * LDSW];
    __shared__ alignas(16) float Rs[CGRP * LDSW];
    __shared__ alignas(16) float St[NWAVES * 16 * STW];

    const int bid = blockIdx.x;          // grid = NB*NH*NG
    const int g   = bid & (NG - 1);
    const int h   = (bid >> 3) % NH;
    const int b   = bid / (NG * NH);

    const int tid = threadIdx.x;

    // ---- Stage 1: b128 coalesced load of 8 channels x 320 cols of L,R rows.
    // Every input element is read from HBM exactly once across the grid.
    const long inBase = ((long)(b * NC + g * CGRP) * NH + h) * NW;  // 16B aligned
    for (int e4 = tid; e4 < (CGRP * NW) / 4; e4 += 256) {
        const int cg = e4 / (NW / 4);
        const int w4 = e4 - cg * (NW / 4);
        *(v4f*)&Ls[cg * LDSW + w4 * 4] =
            *(const v4f*)&left [inBase + (long)cg * (NH * NW) + w4 * 4];
        *(v4f*)&Rs[cg * LDSW + w4 * 4] =
            *(const v4f*)&right[inBase + (long)cg * (NH * NW) + w4 * 4];
    }
    __syncthreads();

    const int wave = tid >> 5;
    const int lane = tid & 31;
    const int lhi  = lane >> 4;   // 0: lanes 0-15, 1: lanes 16-31
    const int lm   = lane & 15;

    float* st = &St[wave * 16 * STW];   // private 16x64(+pad) staging tile
    const long outBase = ((long)((b * NG + g) * NH + h) * NW) * ND;

    // Each wave owns whole 16-wide w-tiles: computes the 5 diagonal M-tiles,
    // re-indexes the band d = 16j + m - n into LDS, then streams out densely.
    for (int i = wave; i < NW / 16; i += NWAVES) {
        const int w0 = i * 16;

        // Zero-fill needed only where d > w can occur (left-edge tiles).
        if (i < 4) {
            #pragma unroll
            for (int c = 0; c < 8; ++c)
                *(v4f*)&st[(2 * c + lhi) * STW + lm * 4] = (v4f){0.f, 0.f, 0.f, 0.f};
        }

        const int jmax = (i < 4) ? i : 4;
        for (int j = 0; j <= jmax; ++j) {
            const int wp0 = (i - j) * 16;      // R rows w' in [wp0, wp0+16)
            const int ar  = w0  + lm;
            const int br  = wp0 + lm;
            // A[m][k]=L[w0+m][k], B[k][n]=R[wp0+n][k]; ISA layout: lanes 0-15
            // hold K=0,1 in VGPR0,1; lanes 16-31 hold K=2,3.
            v2f a0, a1, bb0, bb1;
            a0.x  = Ls[(2 * lhi + 0) * LDSW + ar];
            a0.y  = Ls[(2 * lhi + 1) * LDSW + ar];
            a1.x  = Ls[(2 * lhi + 4) * LDSW + ar];
            a1.y  = Ls[(2 * lhi + 5) * LDSW + ar];
            bb0.x = Rs[(2 * lhi + 0) * LDSW + br];
            bb0.y = Rs[(2 * lhi + 1) * LDSW + br];
            bb1.x = Rs[(2 * lhi + 4) * LDSW + br];
            bb1.y = Rs[(2 * lhi + 5) * LDSW + br];

            v8f c = {};
            c = __builtin_amdgcn_wmma_f32_16x16x4_f32(false, a0, false, bb0,
                                                      (short)0, c, false, false);
            c = __builtin_amdgcn_wmma_f32_16x16x4_f32(false, a1, false, bb1,
                                                      (short)0, c, false, false);

            // D layout: VGPR r -> M = r + 8*lhi, N = lm. Scatter M[m][n] to
            // staging slot [m][d], d = 16j + m - n. Only j==0 / j==4 can fall
            // outside [0,64): clamp those into the 4 dump slots (no exec toggle).
            if ((unsigned)(j - 1) < 3u) {          // j = 1,2,3: always in range
                #pragma unroll
                for (int r = 0; r < 8; ++r) {
                    const int m = r + 8 * lhi;
                    st[m * STW + (16 * j + m - lm)] = c[r];
                }
            } else {
                #pragma unroll
                for (int r = 0; r < 8; ++r) {
                    const int m = r + 8 * lhi;
                    const int d = 16 * j + m - lm;
                    const int idx = ((unsigned)d < (unsigned)ND) ? d : (ND + (lm & 3));
                    st[m * STW + idx] = c[r];
                }
            }
        }

        // Dense readback: each global_store_b128 writes 512 contiguous bytes.
        #pragma unroll
        for (int c = 0; c < 8; ++c) {
            const int m = 2 * c + lhi;
            v4f v = *(const v4f*)&st[m * STW + lm * 4];
            v *= 0.125f;    // group mean (1/Cg), applied once at writeback
            __builtin_nontemporal_store(
                v, (v4f*)&out[outBase + (long)(w0 + m) * ND + lm * 4]);
        }
    }
}

extern "C" void kernel_launch(void* const* d_in, const int* in_sizes, int n_in,
                              void* d_out, int out_size, void* d_ws, size_t ws_size,
                              hipStream_t stream) {
    const float* left  = (const float*)d_in[0];
    const float* right = (const float*)d_in[1];
    float* out = (float*)d_out;

    const int nblocks = NB * NH * NG;   // 5120 blocks, 8 wave32s each
    cost_volume_wmma_kernel<<<nblocks, 256, 0, stream>>>(left, right, out);
}